// Seq2matAttention_21260088115480
// MI455X (gfx1250) — compile-verified
//
#include <hip/hip_runtime.h>
#include <hip/hip_bf16.h>

typedef __attribute__((ext_vector_type(2))) float v2f;
typedef __attribute__((ext_vector_type(8))) float v8f;

#define DD   1024      // D*D
#define LSEQ 2048
#define BB   8
#define MBLK 64
#define NBLK 128
#define KBLK 32

// ---------------------------------------------------------------------------
// Kernel 1: h[m, f] = sum_e X[m, e] * Wd[f, e] + bd[f]
//   M = B*L = 16384, N = K = 1024.  f32 WMMA 16x16x4.
// ---------------------------------------------------------------------------
__global__ __launch_bounds__(256) void dense_gemm_kernel(
    const float* __restrict__ X, const float* __restrict__ Wd,
    const float* __restrict__ bd, float* __restrict__ H)
{
    __shared__ float Xs[MBLK][KBLK];   // 8 KB
    __shared__ float Ws[NBLK][KBLK];   // 16 KB  (stored [n][k] -> B frags contiguous)

    const int tid  = threadIdx.x;
    const int lane = tid & 31;
    const int wave = tid >> 5;       // 0..7
    const int wm   = wave & 3;       // M sub-tile (4 x 16 rows)
    const int wn   = wave >> 2;      // N sub-strip (2 x 64 cols)
    const int m0   = blockIdx.x * MBLK;
    const int n0   = blockIdx.y * NBLK;

    const int lh = lane >> 4;        // 0: K=0,1   1: K=2,3
    const int lm = lane & 15;

    v8f acc[4];
    #pragma unroll
    for (int t = 0; t < 4; ++t) acc[t] = (v8f){0,0,0,0,0,0,0,0};

    for (int k0 = 0; k0 < DD; k0 += KBLK) {
        // Stage X tile: 64x32 f32 = 512 float4, 2 per thread (b128 loads)
        #pragma unroll
        for (int r = 0; r < 2; ++r) {
            int idx  = tid + r * 256;          // 0..511
            int row  = idx >> 3;
            int colv = idx & 7;
            float4 v = *(const float4*)&X[(size_t)(m0 + row) * DD + k0 + colv * 4];
            *(float4*)&Xs[row][colv * 4] = v;
        }
        // Stage W tile: 128x32 f32 = 1024 float4, 4 per thread
        #pragma unroll
        for (int r = 0; r < 4; ++r) {
            int idx  = tid + r * 256;          // 0..1023
            int row  = idx >> 3;
            int colv = idx & 7;
            float4 v = *(const float4*)&Wd[(size_t)(n0 + row) * DD + k0 + colv * 4];
            *(float4*)&Ws[row][colv * 4] = v;
        }
        __syncthreads();

        #pragma unroll
        for (int kk = 0; kk < KBLK; kk += 4) {
            const int ka = kk + lh * 2;
            // A fragment: 16x4, lane lm = row, (lh ? K=2,3 : K=0,1)
            v2f a = *(const v2f*)&Xs[wm * 16 + lm][ka];
            #pragma unroll
            for (int t = 0; t < 4; ++t) {
                // B fragment: B[k][n] = Wd[n][k]; Ws stored [n][k] -> contiguous pair
                v2f b = *(const v2f*)&Ws[wn * 64 + t * 16 + lm][ka];
                acc[t] = __builtin_amdgcn_wmma_f32_16x16x4_f32(
                    false, a, false, b, (short)0, acc[t], false, false);
            }
        }
        __syncthreads();
    }

    // Epilogue: bias + store.  C/D layout: VGPR v, lanes 0-15: M=v, lanes 16-31: M=v+8.
    #pragma unroll
    for (int t = 0; t < 4; ++t) {
        const int col  = n0 + wn * 64 + t * 16 + lm;
        const float bv = bd[col];
        #pragma unroll
        for (int v = 0; v < 8; ++v) {
            const int row = m0 + wm * 16 + v + lh * 8;
            H[(size_t)row * DD + col] = acc[t][v] + bv;
        }
    }
}

// ---------------------------------------------------------------------------
// Kernel 2: windowed attention + 32x32 matrix chain.
//   One block = 4 waves = 4 consecutive positions sharing an 8-matrix window.
// ---------------------------------------------------------------------------
__global__ __launch_bounds__(128) void attn_chain_kernel(
    const float* __restrict__ H, const float* __restrict__ Wa,
    const float* __restrict__ ba, float* __restrict__ Out)
{
    __shared__ float win[8][32][32];   // 32 KB: h_pad window (identity at edges)
    __shared__ float scr[4][32][32];   // 16 KB: per-wave chain scratch

    const int tid  = threadIdx.x;
    const int lane = tid & 31;
    const int wave = tid >> 5;                 // position within block: 0..3
    const int p0   = blockIdx.x * 4;           // flat position b*L + c0
    const int bidx = p0 / LSEQ;
    const int c0   = p0 % LSEQ;                // L % 4 == 0 -> never straddles batches

    // ---- stage window: 8 matrices * 1024 f32 = 2048 float4, 16 per thread ----
    #pragma unroll
    for (int it = 0; it < 16; ++it) {
        const int t  = tid + it * 128;         // float4 index 0..2047
        const int w  = t >> 8;                 // matrix 0..7
        const int o  = (t & 255) * 4;          // float offset in matrix
        const int gw = c0 - 2 + w;             // seq index (identity if OOB)
        float4 v;
        if (gw >= 0 && gw < LSEQ) {
            v = *(const float4*)&H[((size_t)bidx * LSEQ + gw) * DD + o];
        } else {
            const int i = o >> 5, j = o & 31;
            v.x = (i == j)     ? 1.0f : 0.0f;
            v.y = (i == j + 1) ? 1.0f : 0.0f;
            v.z = (i == j + 2) ? 1.0f : 0.0f;
            v.w = (i == j + 3) ? 1.0f : 0.0f;
        }
        ((float4*)win)[t] = v;
    }
    __syncthreads();

    // ---- scores: sc[kp] = sum_{i,j,k} M_k[i][j] * Wa[kp][(i*32+j)*5+k] + ba[kp]
    float part[5] = {0.f, 0.f, 0.f, 0.f, 0.f};
    for (int j = 0; j < 32; ++j) {
        #pragma unroll
        for (int k = 0; k < 5; ++k) {
            const float m = win[wave + k][lane][j];      // i = lane
            const int   e = (lane * 32 + j) * 5 + k;
            #pragma unroll
            for (int kp = 0; kp < 5; ++kp)
                part[kp] = fmaf(m, Wa[kp * (DD * 5) + e], part[kp]);
        }
    }
    float sc[5];
    #pragma unroll
    for (int kp = 0; kp < 5; ++kp) {
        float v = part[kp];
        #pragma unroll
        for (int off = 16; off >= 1; off >>= 1)
            v += __shfl_xor(v, off, 32);
        sc[kp] = v + ba[kp];
    }
    // leaky relu + softmax + scalar product
    float mx = -3.4e38f;
    #pragma unroll
    for (int kp = 0; kp < 5; ++kp) {
        sc[kp] = sc[kp] > 0.f ? sc[kp] : 0.01f * sc[kp];
        mx = fmaxf(mx, sc[kp]);
    }
    float den = 0.f, pnum = 1.f;
    #pragma unroll
    for (int kp = 0; kp < 5; ++kp) {
        const float e = __expf(sc[kp] - mx);
        den += e;
        pnum *= e;
    }
    const float dinv = 1.f / den;
    const float s = pnum * (dinv * dinv) * (dinv * dinv) * dinv;  // prod(alpha_k)

    // ---- chain: T = M0 @ M1 @ M2 @ M3 @ M4, scaled by s on final store ----
    const int lh = lane >> 4;
    const int lm = lane & 15;
    v8f acc[4];

    const float* Aw = &win[wave][0][0];   // stage-1 A = M0
    float*       Sw = &scr[wave][0][0];   // per-wave scratch

    #pragma unroll
    for (int stage = 1; stage <= 4; ++stage) {
        const float* Asrc = (stage == 1) ? Aw : Sw;
        const float* Bsrc = &win[wave + stage][0][0];

        #pragma unroll
        for (int t = 0; t < 4; ++t) acc[t] = (v8f){0,0,0,0,0,0,0,0};

        #pragma unroll
        for (int kk = 0; kk < 32; kk += 4) {
            const int ka = kk + lh * 2;
            #pragma unroll
            for (int ti = 0; ti < 2; ++ti) {
                v2f a = *(const v2f*)&Asrc[(ti * 16 + lm) * 32 + ka];
                #pragma unroll
                for (int tj = 0; tj < 2; ++tj) {
                    const int n = tj * 16 + lm;
                    v2f b;
                    b.x = Bsrc[ka * 32 + n];
                    b.y = Bsrc[(ka + 1) * 32 + n];
                    acc[ti * 2 + tj] = __builtin_amdgcn_wmma_f32_16x16x4_f32(
                        false, a, false, b, (short)0, acc[ti * 2 + tj], false, false);
                }
            }
        }

        if (stage < 4) {
            // all LDS reads above precede these writes in program order;
            // same-wave DS ops are in-order -> safe in-place update
            #pragma unroll
            for (int ti = 0; ti < 2; ++ti)
                #pragma unroll
                for (int tj = 0; tj < 2; ++tj)
                    #pragma unroll
                    for (int v = 0; v < 8; ++v)
                        Sw[(ti * 16 + v + lh * 8) * 32 + tj * 16 + lm] = acc[ti * 2 + tj][v];
        } else {
            float* outp = &Out[((size_t)bidx * LSEQ + c0 + wave) * DD];
            #pragma unroll
            for (int ti = 0; ti < 2; ++ti)
                #pragma unroll
                for (int tj = 0; tj < 2; ++tj)
                    #pragma unroll
                    for (int v = 0; v < 8; ++v)
                        outp[(ti * 16 + v + lh * 8) * 32 + tj * 16 + lm] =
                            s * acc[ti * 2 + tj][v];
        }
    }
}

// ---------------------------------------------------------------------------
extern "C" void kernel_launch(void* const* d_in, const int* in_sizes, int n_in,
                              void* d_out, int out_size, void* d_ws, size_t ws_size,
                              hipStream_t stream) {
    const float* X  = (const float*)d_in[0];   // (B, L, 32, 32) f32
    const float* Wd = (const float*)d_in[1];   // (1024, 1024)
    const float* bd = (const float*)d_in[2];   // (1024,)
    const float* Wa = (const float*)d_in[3];   // (5, 5120)
    const float* ba = (const float*)d_in[4];   // (5,)
    float* H   = (float*)d_ws;                 // h: 16384 x 1024 f32 = 64 MB
    float* Out = (float*)d_out;

    dim3 g1((BB * LSEQ) / MBLK, DD / NBLK);    // 256 x 8 blocks
    dense_gemm_kernel<<<g1, 256, 0, stream>>>(X, Wd, bd, H);

    attn_chain_kernel<<<(BB * LSEQ) / 4, 128, 0, stream>>>(H, Wa, ba, Out);
}